// GCNNClassifier_3324304687691
// MI455X (gfx1250) — compile-verified
//
#include <hip/hip_runtime.h>
#include <hip/hip_bf16.h>

#define N_CH  128
#define N_CLS 16

typedef float v2f __attribute__((ext_vector_type(2)));
typedef float v8f __attribute__((ext_vector_type(8)));

// ---------------- degree / normalization ----------------

__global__ void k_init_deg(float* deg, int N) {
  int i = blockIdx.x * blockDim.x + threadIdx.x;
  if (i < N) deg[i] = 1.0f;                    // self-loop weight
}

__global__ void k_deg_accum(const int* __restrict__ ei, const float* __restrict__ ew,
                            float* deg, int E) {
  int e = blockIdx.x * blockDim.x + threadIdx.x;
  if (e < E) unsafeAtomicAdd(&deg[ei[E + e]], ew[e]);   // dst row of edge_index
}

__global__ void k_dinv(float* deg, int N) {
  int i = blockIdx.x * blockDim.x + threadIdx.x;
  if (i < N) {
    float d = deg[i];
    deg[i] = d > 0.0f ? rsqrtf(d) : 0.0f;
  }
}

// ---------------- h = x @ W1 via V_WMMA_F32_16X16X4_F32 ----------------
// One wave computes a 16x16 tile of h; 8 waves/block cover all 128 columns
// for a 16-row stripe. fp32 WMMA keeps full reference precision.

__global__ void __launch_bounds__(256) k_gemm_xw(const float* __restrict__ x,
                                                const float* __restrict__ W1,
                                                float* __restrict__ h, int N) {
  const int lane = threadIdx.x & 31;
  const int wave = threadIdx.x >> 5;
  const int row0 = blockIdx.x * 16;
  const int col0 = wave * 16;
  const int mn   = lane & 15;            // M for A-frag, N for B/D-frag
  const int koff = (lane >> 4) << 1;     // lanes 0-15 -> K,K+1 ; lanes 16-31 -> K+2,K+3

  int arow = row0 + mn;
  if (arow >= N) arow = N - 1;           // clamp (tail stripe); stores are guarded
  const float* aptr = x  + (size_t)arow * N_CH + koff;
  const float* bptr = W1 + (size_t)koff * N_CH + col0 + mn;

  v8f acc = {0.f, 0.f, 0.f, 0.f, 0.f, 0.f, 0.f, 0.f};
#pragma unroll
  for (int K = 0; K < N_CH; K += 4) {
    v2f a = *(const v2f*)(aptr + K);     // A: 16x4 fp32 fragment
    v2f b;                                // B: 4x16 fp32 fragment
    b.x = bptr[(size_t)K * N_CH];
    b.y = bptr[(size_t)(K + 1) * N_CH];
    acc = __builtin_amdgcn_wmma_f32_16x16x4_f32(
        /*neg_a=*/false, a, /*neg_b=*/false, b,
        /*c_mod=*/(short)0, acc, /*reuse_a=*/false, /*reuse_b=*/false);
  }

  const int rbase = row0 + ((lane >> 4) << 3);   // D: VGPR v -> M = v (+8 for hi lanes)
#pragma unroll
  for (int v = 0; v < 8; ++v) {
    int r = rbase + v;
    if (r < N) h[(size_t)r * N_CH + col0 + mn] = acc[v];
  }
}

// ---------------- aggregation ----------------

__global__ void k_agg_init(const float* __restrict__ h, const float* __restrict__ dinv,
                           const float* __restrict__ b1, float* __restrict__ agg, int N) {
  int i = blockIdx.x;
  int t = threadIdx.x;
  if (i < N) {
    float di = dinv[i];
    agg[(size_t)i * N_CH + t] = h[(size_t)i * N_CH + t] * di * di + b1[t];
  }
}

__global__ void __launch_bounds__(256) k_agg_edges(const int* __restrict__ ei,
                                                   const float* __restrict__ ew,
                                                   const float* __restrict__ dinv,
                                                   const float* __restrict__ h,
                                                   float* __restrict__ agg, int E) {
  int e = blockIdx.x * 8 + (threadIdx.x >> 5);   // one wave32 per edge
  if (e >= E) return;
  int lane = threadIdx.x & 31;
  int src = ei[e];
  int dst = ei[E + e];
  float norm = dinv[src] * ew[e] * dinv[dst];
  const float4 hv = *(const float4*)(h + (size_t)src * N_CH + lane * 4);
  float* ap = agg + (size_t)dst * N_CH + lane * 4;
  unsafeAtomicAdd(ap + 0, hv.x * norm);
  unsafeAtomicAdd(ap + 1, hv.y * norm);
  unsafeAtomicAdd(ap + 2, hv.z * norm);
  unsafeAtomicAdd(ap + 3, hv.w * norm);
}

// ---------------- fused LayerNorm + ReLU + FC (128 -> 16) ----------------

__global__ void __launch_bounds__(256) k_ln_fc(const float* __restrict__ agg,
                                               const float* __restrict__ g,
                                               const float* __restrict__ b,
                                               const float* __restrict__ Wfc,
                                               const float* __restrict__ bfc,
                                               float* __restrict__ out, int N) {
  int node = blockIdx.x * 8 + (threadIdx.x >> 5);  // one wave32 per node
  if (node >= N) return;
  int lane = threadIdx.x & 31;

  float4 v = *(const float4*)(agg + (size_t)node * N_CH + lane * 4);
  float s = v.x + v.y + v.z + v.w;
#pragma unroll
  for (int m = 16; m >= 1; m >>= 1) s += __shfl_xor(s, m, 32);
  float mean = s * (1.0f / 128.0f);

  float dx = v.x - mean, dy = v.y - mean, dz = v.z - mean, dw = v.w - mean;
  float ss = dx * dx + dy * dy + dz * dz + dw * dw;
#pragma unroll
  for (int m = 16; m >= 1; m >>= 1) ss += __shfl_xor(ss, m, 32);
  float rstd = rsqrtf(ss * (1.0f / 128.0f) + 1e-5f);

  float4 g4 = *(const float4*)(g + lane * 4);
  float4 b4 = *(const float4*)(b + lane * 4);
  float hr[4];
  hr[0] = fmaxf(dx * rstd * g4.x + b4.x, 0.0f);
  hr[1] = fmaxf(dy * rstd * g4.y + b4.y, 0.0f);
  hr[2] = fmaxf(dz * rstd * g4.z + b4.z, 0.0f);
  hr[3] = fmaxf(dw * rstd * g4.w + b4.w, 0.0f);

  float p[N_CLS];
#pragma unroll
  for (int j = 0; j < N_CLS; ++j) p[j] = 0.0f;
#pragma unroll
  for (int t = 0; t < 4; ++t) {
    const float* wr = Wfc + (size_t)(lane * 4 + t) * N_CLS;
#pragma unroll
    for (int j = 0; j < N_CLS; ++j) p[j] += hr[t] * wr[j];
  }
#pragma unroll
  for (int j = 0; j < N_CLS; ++j) {
#pragma unroll
    for (int m = 16; m >= 1; m >>= 1) p[j] += __shfl_xor(p[j], m, 32);
  }
  if (lane == 0) {
#pragma unroll
    for (int j = 0; j < N_CLS; ++j) out[(size_t)node * N_CLS + j] = p[j] + bfc[j];
  }
}

// ---------------- launch ----------------

extern "C" void kernel_launch(void* const* d_in, const int* in_sizes, int n_in,
                              void* d_out, int out_size, void* d_ws, size_t ws_size,
                              hipStream_t stream) {
  const float* x   = (const float*)d_in[0];
  const int*   ei  = (const int*)d_in[1];    // edge_index [2,E] flat (harness int mapping)
  const float* ew  = (const float*)d_in[2];
  const float* W1  = (const float*)d_in[3];
  const float* b1  = (const float*)d_in[4];
  const float* lng = (const float*)d_in[5];
  const float* lnb = (const float*)d_in[6];
  const float* Wfc = (const float*)d_in[7];
  const float* bfc = (const float*)d_in[8];
  float* out = (float*)d_out;

  const int N = in_sizes[0] / N_CH;
  const int E = in_sizes[2];

  // workspace layout: deg/dinv [N] | h [N*128] | agg [N*128]
  char*  ws  = (char*)d_ws;
  float* deg = (float*)ws;
  size_t off = (((size_t)N * 4) + 255) & ~(size_t)255;
  float* h   = (float*)(ws + off);
  float* agg = h + (size_t)N * N_CH;

  k_init_deg <<<(N + 255) / 256, 256, 0, stream>>>(deg, N);
  k_deg_accum<<<(E + 255) / 256, 256, 0, stream>>>(ei, ew, deg, E);
  k_dinv     <<<(N + 255) / 256, 256, 0, stream>>>(deg, N);
  k_gemm_xw  <<<(N + 15) / 16, 256, 0, stream>>>(x, W1, h, N);
  k_agg_init <<<N, N_CH, 0, stream>>>(h, deg, b1, agg, N);
  k_agg_edges<<<(E + 7) / 8, 256, 0, stream>>>(ei, ew, deg, h, agg, E);
  k_ln_fc    <<<(N + 7) / 8, 256, 0, stream>>>(agg, lng, lnb, Wfc, bfc, out, N);
}